// RelationAwareDiscriminator_50568944943402
// MI455X (gfx1250) — compile-verified
//
#include <hip/hip_runtime.h>
#include <hip/hip_bf16.h>

// CDNA5 / gfx1250, wave32. fp32 WMMA path (V_WMMA_F32_16X16X4_F32).

typedef __attribute__((ext_vector_type(2))) float v2f;
typedef __attribute__((ext_vector_type(8))) float v8f;

#define NB       512   // batch
#define DIM      128   // embed dim
#define TM       64    // u-rows per workgroup
#define TN       64    // v-cols per inner tile
#define KC       32    // rel k-chunk rows staged in LDS
#define RS       (DIM + 4)  // padded LDS row stride (floats), row-major tiles
#define KS       (KC + 2)   // padded LDS row stride for transposed rel chunk
#define NTHREADS 128   // 4 waves of 32

__global__ __launch_bounds__(NTHREADS)
void rad_wmma_kernel(const int* __restrict__ u_idx,
                     const int* __restrict__ v_idx,
                     const int* __restrict__ r_idx,
                     const float* __restrict__ node_emb,
                     const float* __restrict__ rel_mats,
                     float* __restrict__ out)
{
    __shared__ float rel_t[DIM * KS]; // 17.0 KB : rel chunk, TRANSPOSED [n][k]
    __shared__ float ut_s [TM * RS];  // 33.8 KB : U tile, then (aliased) T tile
    __shared__ float v_s  [TN * RS];  // 33.8 KB : gathered V tile

    const int tid   = threadIdx.x;
    const int wave  = tid >> 5;            // 0..3
    const int lane  = tid & 31;
    const int l16   = lane & 15;
    const int khi   = (lane >> 4) << 1;    // 0 or 2 (A/B frag K select)
    const int rowhi = (lane >> 4) << 3;    // 0 or 8 (C/D frag M select)

    const int utile = blockIdx.x & 7;      // 8 u-tiles of 64 rows
    const int b     = blockIdx.x >> 3;     // 0..511

    const float* relb = rel_mats + (size_t)r_idx[b] * (DIM * DIM);

    // ---------------- load gathered U tile into ut_s ----------------
    for (int i = 0; i < (TM * DIM / 4) / NTHREADS; ++i) {   // 16 iters
        int q   = i * NTHREADS + tid;                       // 0..2047 float4s
        int row = q >> 5;                                   // 32 float4 / row
        int c4  = (q & 31) << 2;
        float4 val = *(const float4*)(node_emb +
                        (size_t)u_idx[utile * TM + row] * DIM + c4);
        *(float4*)&ut_s[row * RS + c4] = val;               // RS*4 % 16 == 0
    }

    // ---------------- Stage A: T = U_tile @ rel[b] ----------------
    // wave owns a 16-row stripe (m = wave*16 ..), all 8 n-tiles of rel cols.
    {
        v8f accA[DIM / 16] = {};                            // 8 tiles, 64 VGPRs
        const int mrow = wave * 16 + l16;

        for (int kc = 0; kc < DIM / KC; ++kc) {             // 4 chunks of K=32
            __syncthreads();                                // rel_t reads done
            // load rel rows [kc*KC, kc*KC+KC), transposing into rel_t[n][k]
            for (int i = 0; i < (KC * DIM / 4) / NTHREADS; ++i) { // 8 iters
                int q   = i * NTHREADS + tid;               // 0..1023 float4s
                int kro = q >> 5;                           // k row in chunk
                int c4  = (q & 31) << 2;                    // n base
                float4 val = *(const float4*)(relb +
                                (size_t)(kc * KC + kro) * DIM + c4);
                rel_t[(c4 + 0) * KS + kro] = val.x;
                rel_t[(c4 + 1) * KS + kro] = val.y;
                rel_t[(c4 + 2) * KS + kro] = val.z;
                rel_t[(c4 + 3) * KS + kro] = val.w;
            }
            // prefetch next chunk (global_prefetch_b8), one 128B line/thread
            if (kc + 1 < DIM / KC)
                __builtin_prefetch(relb + (size_t)(kc + 1) * KC * DIM + tid * 32,
                                   0, 0);
            __syncthreads();                                // ut_s + rel_t ready

            for (int kk = 0; kk < KC; kk += 4) {
                // A frag: U[m][k..k+3], contiguous k-pair -> ds b64
                v2f a = *(const v2f*)&ut_s[mrow * RS + kc * KC + kk + khi];
                #pragma unroll
                for (int j = 0; j < DIM / 16; ++j) {
                    // B frag: rel_t[n][k], contiguous k-pair -> ds b64
                    v2f bb = *(const v2f*)
                        &rel_t[(j * 16 + l16) * KS + kk + khi];
                    accA[j] = __builtin_amdgcn_wmma_f32_16x16x4_f32(
                        false, a, false, bb, (short)0, accA[j], false, false);
                }
            }
        }

        __syncthreads();                 // all U reads done -> alias ut_s as T
        #pragma unroll
        for (int j = 0; j < DIM / 16; ++j)
            #pragma unroll
            for (int i = 0; i < 8; ++i)
                ut_s[(wave * 16 + rowhi + i) * RS + j * 16 + l16] = accA[j][i];
    }
    __syncthreads();                     // T tile visible to all waves

    // ---------------- Stage B: S = T @ V_tile^T, sigmoid, store ----------------
    const int mrow = wave * 16 + l16;
    for (int vt = 0; vt < NB / TN; ++vt) {                  // 8 v-tiles
        if (vt) __syncthreads();                            // v_s reads done
        for (int i = 0; i < (TN * DIM / 4) / NTHREADS; ++i) { // 16 iters
            int q   = i * NTHREADS + tid;
            int row = q >> 5;
            int c4  = (q & 31) << 2;
            float4 val = *(const float4*)(node_emb +
                            (size_t)v_idx[vt * TN + row] * DIM + c4);
            *(float4*)&v_s[row * RS + c4] = val;
        }
        // prefetch next gathered V tile rows (two 256B halves per row)
        if (vt + 1 < NB / TN) {
            int prow = tid >> 1;                            // 0..63
            __builtin_prefetch(node_emb +
                (size_t)v_idx[(vt + 1) * TN + prow] * DIM + (tid & 1) * 64,
                0, 0);
        }
        __syncthreads();

        v8f accB[TN / 16] = {};                             // 4 tiles
        for (int kk = 0; kk < DIM; kk += 4) {
            v2f a = *(const v2f*)&ut_s[mrow * RS + kk + khi];
            #pragma unroll
            for (int j = 0; j < TN / 16; ++j) {
                // B = V^T : B[k][n] = V[n][k]; (k,k+1) contiguous in a V row
                v2f bb = *(const v2f*)&v_s[(j * 16 + l16) * RS + kk + khi];
                accB[j] = __builtin_amdgcn_wmma_f32_16x16x4_f32(
                    false, a, false, bb, (short)0, accB[j], false, false);
            }
        }

        // sigmoid + coalesced store: out[b][u][v]
        #pragma unroll
        for (int j = 0; j < TN / 16; ++j) {
            #pragma unroll
            for (int i = 0; i < 8; ++i) {
                int urow = utile * TM + wave * 16 + rowhi + i;
                int vcol = vt * TN + j * 16 + l16;
                float x = accB[j][i];
                out[((size_t)b * NB + urow) * NB + vcol] =
                    1.0f / (1.0f + __expf(-x));
            }
        }
    }
}

extern "C" void kernel_launch(void* const* d_in, const int* in_sizes, int n_in,
                              void* d_out, int out_size, void* d_ws, size_t ws_size,
                              hipStream_t stream) {
    (void)in_sizes; (void)n_in; (void)d_ws; (void)ws_size; (void)out_size;
    const int*   u        = (const int*)d_in[0];
    const int*   v        = (const int*)d_in[1];
    const int*   r        = (const int*)d_in[2];
    const float* node_emb = (const float*)d_in[3];
    const float* rel_mats = (const float*)d_in[4];
    float*       out      = (float*)d_out;

    // 8 u-tiles x 512 batch entries = 4096 workgroups, 128 threads (4 waves)
    rad_wmma_kernel<<<dim3((NB / TM) * NB), dim3(NTHREADS), 0, stream>>>(
        u, v, r, node_emb, rel_mats, out);
}